// Contour_to_mask_80839874445330
// MI455X (gfx1250) — compile-verified
//
#include <hip/hip_runtime.h>

// Contour-to-mask winding-angle kernel for gfx1250 (MI455X).
//
// Strategy: the four per-(pixel,edge) bilinear forms (dot, cross, |d|^2, |r|^2)
// are rank-4 products of pixel features [1, px, py, px^2+py^2] and edge
// features, so a wave computes a 16-pixel x 16-edge tile with four
// V_WMMA_F32_16X16X4_F32 ops, then runs the transcendental epilogue
// (v_rsq + NR, acos, v_tanh_f32) on the f32 accumulators. Coordinates are
// shifted to the wave's pixel-tile center to avoid cancellation in the
// expanded bilinear forms (px' == 0 exactly, |py'| <= 15/1024).

#define SIZE   512
#define NPTS   128
#define KSTEEP 100000.0f
#define EPSC   1e-5f

typedef __attribute__((ext_vector_type(2))) float v2f;
typedef __attribute__((ext_vector_type(8))) float v8f;

static __device__ inline float fast_tanh(float x) {
#if __has_builtin(__builtin_amdgcn_tanhf)
  return __builtin_amdgcn_tanhf(x);          // gfx1250 v_tanh_f32 (1 trans op)
#elif __has_builtin(__builtin_amdgcn_tanh_f32)
  return __builtin_amdgcn_tanh_f32(x);
#else
  return tanhf(x);
#endif
}

static __device__ inline float fast_rsqrt(float x) {
#if __has_builtin(__builtin_amdgcn_rsqf)
  float y = __builtin_amdgcn_rsqf(x);        // v_rsq_f32
#else
  float y = 1.0f / sqrtf(x);
#endif
  y = y * (1.5f - 0.5f * x * y * y);         // one Newton step -> ~full fp32
  return y;
}

__global__ __launch_bounds__(256) void contour_mask_wmma(
    const float* __restrict__ contour, float* __restrict__ out) {
  __shared__ float lx[NPTS + 1];
  __shared__ float ly[NPTS + 1];

  const int tid = threadIdx.x;
  if (tid < NPTS) { lx[tid] = contour[2 * tid]; ly[tid] = contour[2 * tid + 1]; }
  if (tid == NPTS) { lx[NPTS] = contour[0]; ly[NPTS] = contour[1]; }
  __syncthreads();

  const int lane = tid & 31;
  const int wave = tid >> 5;
  const int p0   = (blockIdx.x * 8 + wave) * 16;   // wave's 16 consecutive pixels
  const int i0   = p0 >> 9;                        // grid row  -> px = i0/512
  const int i1b  = p0 & 511;                       // grid cols -> py = (i1b+m)/512

  const float inv_size = 1.0f / (float)SIZE;
  const float ox = (float)i0 * inv_size;                 // shift so px' == 0
  const float oy = ((float)i1b + 7.5f) * inv_size;       // center of 16 py's

  const int  m  = lane & 15;    // A: pixel row  M=m ; B: edge column N=m
  const bool hi = lane >= 16;   // upper half-wave holds K=2,3 (layout mirror of A)

  const float pyl = ((float)m - 7.5f) * inv_size;  // local py' for pixel row m

  // A operand (16x4 f32, 2 VGPRs): features [K0=1, K1=px'(=0), K2=py', K3=px'^2+py'^2]
  v2f A;
  A.x = hi ? pyl : 1.0f;               // VGPR0: lanes0-15 K0, lanes16-31 K2
  A.y = hi ? (pyl * pyl) : 0.0f;       // VGPR1: lanes0-15 K1, lanes16-31 K3

  float acc[8];
#pragma unroll
  for (int r = 0; r < 8; ++r) acc[r] = 0.0f;

  const v8f zero = {0.f, 0.f, 0.f, 0.f, 0.f, 0.f, 0.f, 0.f};

#pragma unroll
  for (int t = 0; t < NPTS / 16; ++t) {
    const int j = t * 16 + m;
    const float ax = lx[j] - ox,     ay = ly[j] - oy;
    const float bx = lx[j + 1] - ox, by = ly[j + 1] - oy;

    // Edge features for the four bilinear forms (coeffs of [1, px, py, quad]).
    const float d0 = ax * bx + ay * by, d1 = -(ax + bx), d2 = -(ay + by), d3 = 1.0f;
    const float c0 = ay * bx - ax * by, c1 = by - ay,    c2 = ax - bx,    c3 = 0.0f;
    const float e0 = ax * ax + ay * ay, e1 = -2.0f * ax, e2 = -2.0f * ay, e3 = 1.0f;
    const float f0 = bx * bx + by * by, f1 = -2.0f * bx, f2 = -2.0f * by, f3 = 1.0f;

#if __has_builtin(__builtin_amdgcn_wmma_f32_16x16x4_f32)
    // B operand (4x16 f32, 2 VGPRs): assume K-striping mirrors A (K0,K2 | K1,K3).
    v2f Bd, Bc, Be, Bf;
    Bd.x = hi ? d2 : d0;  Bd.y = hi ? d3 : d1;
    Bc.x = hi ? c2 : c0;  Bc.y = hi ? c3 : c1;
    Be.x = hi ? e2 : e0;  Be.y = hi ? e3 : e1;
    Bf.x = hi ? f2 : f0;  Bf.y = hi ? f3 : f1;

    v8f Dd = __builtin_amdgcn_wmma_f32_16x16x4_f32(false, A, false, Bd, (short)0, zero, false, false);
    v8f Dc = __builtin_amdgcn_wmma_f32_16x16x4_f32(false, A, false, Bc, (short)0, zero, false, false);
    v8f Dn = __builtin_amdgcn_wmma_f32_16x16x4_f32(false, A, false, Be, (short)0, zero, false, false);
    v8f Dr = __builtin_amdgcn_wmma_f32_16x16x4_f32(false, A, false, Bf, (short)0, zero, false, false);
#else
    // Scalar fallback with identical math + C-layout (lane = edge col, VGPR r = pixel row)
    v8f Dd, Dc, Dn, Dr;
#pragma unroll
    for (int r = 0; r < 8; ++r) {
      const int   mrow = hi ? (r + 8) : r;
      const float py   = ((float)mrow - 7.5f) * inv_size;
      const float q    = py * py;
      Dd[r] = d0 + py * d2 + q * d3;
      Dc[r] = c0 + py * c2 + q * c3;
      Dn[r] = e0 + py * e2 + q * e3;
      Dr[r] = f0 + py * f2 + q * f3;
      (void)d1; (void)c1; (void)e1; (void)f1;  // px' == 0 terms
    }
#endif

    // Transcendental epilogue per (pixel,edge) element.
#pragma unroll
    for (int r = 0; r < 8; ++r) {
      const float n2  = Dn[r] * Dr[r];
      float cosang    = Dd[r] * fast_rsqrt(n2);
      cosang          = fminf(fmaxf(cosang, -1.0f + EPSC), 1.0f - EPSC);
      const float ang = acosf(cosang);
      const float sgn = fast_tanh(KSTEEP * Dc[r]);
      acc[r] = fmaf(sgn, ang, acc[r]);
    }
  }

  // Reduce over edges: sum across lanes within each 16-lane half (C-matrix N-striping).
  const float inv2pi = 0.15915494309189535f;
#pragma unroll
  for (int r = 0; r < 8; ++r) {
    float v = acc[r];
    v += __shfl_xor(v, 1, 32);
    v += __shfl_xor(v, 2, 32);
    v += __shfl_xor(v, 4, 32);
    v += __shfl_xor(v, 8, 32);
    if ((lane & 15) == 0) {
      const int mrow = hi ? (r + 8) : r;
      out[p0 + mrow] = fminf(fmaxf(v * inv2pi, 0.0f), 1.0f);
    }
  }
}

extern "C" void kernel_launch(void* const* d_in, const int* in_sizes, int n_in,
                              void* d_out, int out_size, void* d_ws, size_t ws_size,
                              hipStream_t stream) {
  (void)in_sizes; (void)n_in; (void)d_ws; (void)ws_size; (void)out_size;
  const float* contour = (const float*)d_in[0];
  float* out = (float*)d_out;
  // 262144 pixels / (8 waves * 16 pixels) = 2048 blocks of 256 threads.
  contour_mask_wmma<<<dim3(2048), dim3(256), 0, stream>>>(contour, out);
}